// RiskSensitiveGNN_51058571214865
// MI455X (gfx1250) — compile-verified
//
#include <hip/hip_runtime.h>
#include <hip/hip_bf16.h>
#include <math.h>

// ---------------------------------------------------------------------------
// CDNA5 (gfx1250) implementation of RiskSensitiveGNN.
// All GEMMs use V_WMMA_F32_16X16X4_F32 (full fp32 matrix op, wave32):
//   A 16x4 f32: lanes 0-15 -> M=lane, VGPR0=K0,VGPR1=K1; lanes 16-31 -> K2,K3
//   B 4x16 f32: VGPR0 lanes0-15 = row K0, lanes16-31 = row K2; VGPR1 = K1/K3
//   C/D 16x16 : VGPR v, lanes0-15 -> (M=v, N=lane); lanes16-31 -> (M=v+8)
// ---------------------------------------------------------------------------

typedef float v2f __attribute__((ext_vector_type(2)));
typedef float v8f __attribute__((ext_vector_type(8)));

#define HID 64
#define EHID 32
#define HEADS 4

// ---------------- node encoder: Lin(9,32)+ReLU+Lin(32,64)+LN(64) -----------
__global__ void k_node_enc(const float* __restrict__ x,
                           const float* __restrict__ w1, const float* __restrict__ b1,
                           const float* __restrict__ w2, const float* __restrict__ b2,
                           const float* __restrict__ g, const float* __restrict__ beta,
                           float* __restrict__ h, int N)
{
    __shared__ float sx[9];
    __shared__ float shid[32];
    __shared__ float sred[64];
    int n = blockIdx.x;
    int t = threadIdx.x;                 // 64 threads
    if (t < 9) sx[t] = x[n * 9 + t];
    __syncthreads();
    if (t < 32) {
        float acc = b1[t];
        for (int i = 0; i < 9; ++i) acc = fmaf(sx[i], w1[i * 32 + t], acc);
        shid[t] = acc > 0.f ? acc : 0.f;
    }
    __syncthreads();
    float o = b2[t];
    for (int j = 0; j < 32; ++j) o = fmaf(shid[j], w2[j * 64 + t], o);
    sred[t] = o; __syncthreads();
    for (int s = 32; s > 0; s >>= 1) { if (t < s) sred[t] += sred[t + s]; __syncthreads(); }
    float mean = sred[0] * (1.f / 64.f); __syncthreads();
    float d = o - mean;
    sred[t] = d * d; __syncthreads();
    for (int s = 32; s > 0; s >>= 1) { if (t < s) sred[t] += sred[t + s]; __syncthreads(); }
    float var = sred[0] * (1.f / 64.f);
    h[n * 64 + t] = d * rsqrtf(var + 1e-5f) * g[t] + beta[t];
}

// ---------------- edge encoder: Lin(4,16)+ReLU+Lin(16,32)+LN(32) -----------
__global__ void k_edge_enc(const float* __restrict__ ea,
                           const float* __restrict__ w1, const float* __restrict__ b1,
                           const float* __restrict__ w2, const float* __restrict__ b2,
                           const float* __restrict__ g, const float* __restrict__ beta,
                           float* __restrict__ ef, int E)
{
    __shared__ float sw1[64], sb1[16], sw2[512], sb2[32], sg[32], sbeta[32];
    int t = threadIdx.x;                 // 256 threads
    for (int i = t; i < 64; i += 256) sw1[i] = w1[i];
    if (t < 16) sb1[t] = b1[t];
    for (int i = t; i < 512; i += 256) sw2[i] = w2[i];
    if (t < 32) { sb2[t] = b2[t]; sg[t] = g[t]; sbeta[t] = beta[t]; }
    __syncthreads();
    int e = blockIdx.x * 256 + t;
    if (e >= E) return;
    float a0 = ea[e * 4 + 0], a1 = ea[e * 4 + 1], a2 = ea[e * 4 + 2], a3 = ea[e * 4 + 3];
    float hid[16];
    for (int j = 0; j < 16; ++j) {
        float acc = sb1[j] + a0 * sw1[j] + a1 * sw1[16 + j] + a2 * sw1[32 + j] + a3 * sw1[48 + j];
        hid[j] = acc > 0.f ? acc : 0.f;
    }
    float out[32]; float mean = 0.f;
    for (int k = 0; k < 32; ++k) {
        float acc = sb2[k];
        for (int j = 0; j < 16; ++j) acc = fmaf(hid[j], sw2[j * 32 + k], acc);
        out[k] = acc; mean += acc;
    }
    mean *= (1.f / 32.f);
    float var = 0.f;
    for (int k = 0; k < 32; ++k) { float d = out[k] - mean; var += d * d; }
    var *= (1.f / 32.f);
    float inv = rsqrtf(var + 1e-5f);
    for (int k = 0; k < 32; ++k) ef[e * 32 + k] = (out[k] - mean) * inv * sg[k] + sbeta[k];
}

// ---------------- risk attention (WMMA): [16edges,96]@[96,64] tanh -> dot64 sigmoid
__global__ void k_risk(const float* __restrict__ h, const float* __restrict__ ef,
                       const int* __restrict__ ei,            // row = ei[e]
                       const float* __restrict__ w1, const float* __restrict__ b1,
                       const float* __restrict__ w2, const float* __restrict__ b2,
                       float* __restrict__ rw, int E)
{
    __shared__ float sA[16 * 96];
    __shared__ float sT[16 * 64];
    __shared__ int   srow[16];
    int t = threadIdx.x;                 // 64 threads = 2 waves
    int e0 = blockIdx.x * 16;
    if (t < 16) srow[t] = ei[e0 + t];
    __syncthreads();
    for (int idx = t; idx < 16 * 96; idx += 64) {
        int m = idx / 96, c = idx % 96;
        sA[idx] = (c < 64) ? h[srow[m] * 64 + c] : ef[(e0 + m) * 32 + (c - 64)];
    }
    __syncthreads();
    int wave = t >> 5, lane = t & 31, half = lane >> 4, lr = lane & 15;
    for (int tt = 0; tt < 2; ++tt) {
        int n0 = wave * 32 + tt * 16;
        v8f c = {};
        for (int k0 = 0; k0 < 96; k0 += 4) {
            v2f a, b;
            a.x = sA[lr * 96 + k0 + 2 * half + 0];
            a.y = sA[lr * 96 + k0 + 2 * half + 1];
            b.x = w1[(k0 + 2 * half + 0) * 64 + n0 + lr];
            b.y = w1[(k0 + 2 * half + 1) * 64 + n0 + lr];
            c = __builtin_amdgcn_wmma_f32_16x16x4_f32(false, a, false, b, (short)0, c, false, false);
        }
        for (int v = 0; v < 8; ++v) {
            int row = v + 8 * half, col = n0 + lr;
            sT[row * 64 + col] = tanhf(c[v] + b1[col]);
        }
    }
    __syncthreads();
    if (t < 16) {
        float s = b2[0];
        for (int j = 0; j < 64; ++j) s = fmaf(sT[t * 64 + j], w2[j], s);
        rw[e0 + t] = 1.f / (1.f + expf(-s));
    }
}

// ---------------- xproj (WMMA) + per-node attention scalars si/sj ----------
__global__ void k_xproj(const float* __restrict__ xh, const float* __restrict__ wl, // [64][256]
                        const float* __restrict__ wa,                               // [160]
                        float* __restrict__ xproj, float* __restrict__ sij, int N)
{
    __shared__ float sA[16 * 64];
    __shared__ float sP[16 * 256];
    int t = threadIdx.x;                 // 256 threads = 8 waves
    int m0 = blockIdx.x * 16;
    for (int idx = t; idx < 16 * 64; idx += 256) sA[idx] = xh[m0 * 64 + idx];
    __syncthreads();
    int wave = t >> 5, lane = t & 31, half = lane >> 4, lr = lane & 15;
    for (int tt = 0; tt < 2; ++tt) {
        int n0 = wave * 32 + tt * 16;
        v8f c = {};
        for (int k0 = 0; k0 < 64; k0 += 4) {
            v2f a, b;
            a.x = sA[lr * 64 + k0 + 2 * half + 0];
            a.y = sA[lr * 64 + k0 + 2 * half + 1];
            b.x = wl[(k0 + 2 * half + 0) * 256 + n0 + lr];
            b.y = wl[(k0 + 2 * half + 1) * 256 + n0 + lr];
            c = __builtin_amdgcn_wmma_f32_16x16x4_f32(false, a, false, b, (short)0, c, false, false);
        }
        for (int v = 0; v < 8; ++v) {
            int row = v + 8 * half, col = n0 + lr;
            float val = c[v];
            xproj[(m0 + row) * 256 + col] = val;
            sP[row * 256 + col] = val;
        }
    }
    __syncthreads();
    if (t < 128) {                       // 16 nodes * 4 heads * {si,sj}
        int m = t >> 3, q = t & 7, w = q & 1, hh = q >> 1;
        const float* wv = wa + w * 64;   // wa_i at 0, wa_j at 64
        float s = 0.f;
        for (int d = 0; d < 64; ++d) s = fmaf(sP[m * 256 + hh * 64 + d], wv[d], s);
        sij[(m0 + m) * 8 + w * 4 + hh] = s;
    }
}

// ---------------- per-layer init: zero agg, reset amax/asum ----------------
__global__ void k_init(float* __restrict__ agg, float* __restrict__ amax,
                       float* __restrict__ asum, int total)
{
    int i = blockIdx.x * 256 + threadIdx.x;
    if (i < total) agg[i] = 0.f;
    if (i < 4) { amax[i] = -3.0e38f; asum[i] = 0.f; }
}

__device__ inline void atomicMaxF(float* addr, float val)
{
    int* ai = (int*)addr;
    int old = __float_as_int(*addr);
    while (__int_as_float(old) < val) {
        int assumed = old;
        old = atomicCAS(ai, assumed, __float_as_int(val));
        if (old == assumed) break;
    }
}

// ---------------- alpha = leaky(si+sj+se)*rw, global per-head max ----------
__global__ void k_alpha(const float* __restrict__ ef, const int* __restrict__ ei,
                        const float* __restrict__ sij, const float* __restrict__ rw,
                        const float* __restrict__ wa, float* __restrict__ palpha,
                        float* __restrict__ amax, int E)
{
    __shared__ float swa[160];
    __shared__ float wred[8 * 4];
    int t = threadIdx.x;                 // 256 threads
    if (t < 160) swa[t] = wa[t];
    __syncthreads();
    int e = blockIdx.x * 256 + t;
    float a[4] = { -3.0e38f, -3.0e38f, -3.0e38f, -3.0e38f };
    if (e < E) {
        int r = ei[e], c = ei[E + e];
        float se = 0.f;
        for (int j = 0; j < 32; ++j) se = fmaf(ef[e * 32 + j], swa[128 + j], se);
        float w = rw[e];
        for (int hh = 0; hh < 4; ++hh) {
            float v = sij[r * 8 + hh] + sij[c * 8 + 4 + hh] + se;
            v = v >= 0.f ? v : 0.2f * v;   // leaky_relu(0.2)
            v *= w;                        // risk modulation
            palpha[e * 4 + hh] = v;
            a[hh] = v;
        }
    }
    for (int hh = 0; hh < 4; ++hh) {
        float v = a[hh];
        for (int off = 16; off > 0; off >>= 1) v = fmaxf(v, __shfl_xor(v, off, 32));
        a[hh] = v;
    }
    int lane = t & 31, wv = t >> 5;
    if (lane == 0) for (int hh = 0; hh < 4; ++hh) wred[wv * 4 + hh] = a[hh];
    __syncthreads();
    if (t < 4) {
        float v = wred[t];
        for (int wi = 1; wi < 8; ++wi) v = fmaxf(v, wred[wi * 4 + t]);
        atomicMaxF(&amax[t], v);
    }
}

// ---------------- p = exp(alpha - max); global per-head sum ----------------
__global__ void k_expsum(float* __restrict__ palpha, const float* __restrict__ amax,
                         float* __restrict__ asum, int E)
{
    __shared__ float wred[8 * 4];
    int t = threadIdx.x;
    int e = blockIdx.x * 256 + t;
    float a[4] = { 0.f, 0.f, 0.f, 0.f };
    if (e < E) {
        for (int hh = 0; hh < 4; ++hh) {
            float p = expf(palpha[e * 4 + hh] - amax[hh]);
            palpha[e * 4 + hh] = p;
            a[hh] = p;
        }
    }
    for (int hh = 0; hh < 4; ++hh) {
        float v = a[hh];
        for (int off = 16; off > 0; off >>= 1) v += __shfl_xor(v, off, 32);
        a[hh] = v;
    }
    int lane = t & 31, wv = t >> 5;
    if (lane == 0) for (int hh = 0; hh < 4; ++hh) wred[wv * 4 + hh] = a[hh];
    __syncthreads();
    if (t < 4) {
        float v = 0.f;
        for (int wi = 0; wi < 8; ++wi) v += wred[wi * 4 + t];
        atomicAdd(&asum[t], v);
    }
}

// ---------------- aggregation: fused emsg WMMA + gather + scatter-add ------
// block = 128 threads = 4 waves; wave w owns head w (cols w*64 .. w*64+63)
__global__ void k_agg(const float* __restrict__ ef, const int* __restrict__ ei,
                      const float* __restrict__ palpha, const float* __restrict__ asum,
                      const float* __restrict__ we,            // [32][256]
                      const float* __restrict__ xproj,
                      float* __restrict__ agg, int E)
{
    __shared__ float sef[16 * 32];
    __shared__ int   scol[16];
    __shared__ float sw[16 * 4];
    int t = threadIdx.x;
    int e0 = blockIdx.x * 16;
    if (t < 16) scol[t] = ei[E + e0 + t];
    if (t < 64) sw[t] = palpha[e0 * 4 + t] / asum[t & 3];
    for (int idx = t; idx < 512; idx += 128) sef[idx] = ef[e0 * 32 + idx];
    __syncthreads();
    int wave = t >> 5, lane = t & 31, half = lane >> 4, lr = lane & 15;
    for (int tt = 0; tt < 4; ++tt) {
        int n0 = wave * 64 + tt * 16;
        v8f c = {};
        for (int k0 = 0; k0 < 32; k0 += 4) {
            v2f a, b;
            a.x = sef[lr * 32 + k0 + 2 * half + 0];
            a.y = sef[lr * 32 + k0 + 2 * half + 1];
            b.x = we[(k0 + 2 * half + 0) * 256 + n0 + lr];
            b.y = we[(k0 + 2 * half + 1) * 256 + n0 + lr];
            c = __builtin_amdgcn_wmma_f32_16x16x4_f32(false, a, false, b, (short)0, c, false, false);
        }
        int col = n0 + lr;
        for (int v = 0; v < 8; ++v) {
            int m = v + 8 * half;              // edge within tile
            int cn = scol[m];
            float xj = xproj[cn * 256 + col];
            float val = sw[m * 4 + wave] * (xj + c[v]);
            atomicAdd(&agg[cn * 256 + col], val);
        }
    }
}

// ---------------- output transform: relu(mean_h(agg) @ wo + wob) -----------
__global__ void k_otrans(const float* __restrict__ agg, const float* __restrict__ wo,
                         const float* __restrict__ wob, float* __restrict__ xh, int N)
{
    __shared__ float smean[64];
    int n = blockIdx.x, t = threadIdx.x;     // 64 threads
    smean[t] = 0.25f * (agg[n * 256 + t] + agg[n * 256 + 64 + t] +
                        agg[n * 256 + 128 + t] + agg[n * 256 + 192 + t]);
    __syncthreads();
    float acc = wob[t];
    for (int j = 0; j < 64; ++j) acc = fmaf(smean[j], wo[j * 64 + t], acc);
    xh[n * 64 + t] = acc > 0.f ? acc : 0.f;
}

// ---------------- output MLP stage 1 (WMMA): relu([N,64]@[64,128]+b) -------
__global__ void k_out1(const float* __restrict__ xh, const float* __restrict__ w,
                       const float* __restrict__ b, float* __restrict__ hid1, int N)
{
    __shared__ float sA[16 * 64];
    int t = threadIdx.x;                     // 128 threads = 4 waves
    int m0 = blockIdx.x * 16;
    for (int idx = t; idx < 1024; idx += 128) sA[idx] = xh[m0 * 64 + idx];
    __syncthreads();
    int wave = t >> 5, lane = t & 31, half = lane >> 4, lr = lane & 15;
    for (int tt = 0; tt < 2; ++tt) {
        int n0 = wave * 32 + tt * 16;
        v8f c = {};
        for (int k0 = 0; k0 < 64; k0 += 4) {
            v2f a, bb;
            a.x = sA[lr * 64 + k0 + 2 * half + 0];
            a.y = sA[lr * 64 + k0 + 2 * half + 1];
            bb.x = w[(k0 + 2 * half + 0) * 128 + n0 + lr];
            bb.y = w[(k0 + 2 * half + 1) * 128 + n0 + lr];
            c = __builtin_amdgcn_wmma_f32_16x16x4_f32(false, a, false, bb, (short)0, c, false, false);
        }
        for (int v = 0; v < 8; ++v) {
            int row = v + 8 * half, col = n0 + lr;
            float val = c[v] + b[col];
            hid1[(m0 + row) * 128 + col] = val > 0.f ? val : 0.f;
        }
    }
}

// ---------------- output MLP stage 2 (WMMA): [N,128]@[128,256]+b -----------
__global__ void k_out2(const float* __restrict__ hid1, const float* __restrict__ w,
                       const float* __restrict__ b, float* __restrict__ pre, int N)
{
    __shared__ float sA[16 * 128];
    int t = threadIdx.x;                     // 256 threads = 8 waves
    int m0 = blockIdx.x * 16;
    for (int idx = t; idx < 2048; idx += 256) sA[idx] = hid1[m0 * 128 + idx];
    __syncthreads();
    int wave = t >> 5, lane = t & 31, half = lane >> 4, lr = lane & 15;
    for (int tt = 0; tt < 2; ++tt) {
        int n0 = wave * 32 + tt * 16;
        v8f c = {};
        for (int k0 = 0; k0 < 128; k0 += 4) {
            v2f a, bb;
            a.x = sA[lr * 128 + k0 + 2 * half + 0];
            a.y = sA[lr * 128 + k0 + 2 * half + 1];
            bb.x = w[(k0 + 2 * half + 0) * 256 + n0 + lr];
            bb.y = w[(k0 + 2 * half + 1) * 256 + n0 + lr];
            c = __builtin_amdgcn_wmma_f32_16x16x4_f32(false, a, false, bb, (short)0, c, false, false);
        }
        for (int v = 0; v < 8; ++v) {
            int row = v + 8 * half, col = n0 + lr;
            pre[(m0 + row) * 256 + col] = c[v] + b[col];
        }
    }
}

// ---------------- LayerNorm(256) -> d_out ----------------------------------
__global__ void k_ln256(const float* __restrict__ pre, const float* __restrict__ g,
                        const float* __restrict__ beta, float* __restrict__ out, int N)
{
    __shared__ float sred[256];
    int n = blockIdx.x, t = threadIdx.x;     // 256 threads
    float v = pre[n * 256 + t];
    sred[t] = v; __syncthreads();
    for (int s = 128; s > 0; s >>= 1) { if (t < s) sred[t] += sred[t + s]; __syncthreads(); }
    float mean = sred[0] * (1.f / 256.f); __syncthreads();
    float d = v - mean;
    sred[t] = d * d; __syncthreads();
    for (int s = 128; s > 0; s >>= 1) { if (t < s) sred[t] += sred[t + s]; __syncthreads(); }
    float var = sred[0] * (1.f / 256.f);
    out[n * 256 + t] = d * rsqrtf(var + 1e-5f) * g[t] + beta[t];
}

// ---------------------------------------------------------------------------
extern "C" void kernel_launch(void* const* d_in, const int* in_sizes, int n_in,
                              void* d_out, int out_size, void* d_ws, size_t ws_size,
                              hipStream_t stream)
{
    const float* x          = (const float*)d_in[0];
    const float* edge_attr  = (const float*)d_in[1];
    const int*   edge_index = (const int*)  d_in[2];
    const float* ne_w1 = (const float*)d_in[3];  const float* ne_b1 = (const float*)d_in[4];
    const float* ne_w2 = (const float*)d_in[5];  const float* ne_b2 = (const float*)d_in[6];
    const float* ne_g  = (const float*)d_in[7];  const float* ne_be = (const float*)d_in[8];
    const float* ee_w1 = (const float*)d_in[9];  const float* ee_b1 = (const float*)d_in[10];
    const float* ee_w2 = (const float*)d_in[11]; const float* ee_b2 = (const float*)d_in[12];
    const float* ee_g  = (const float*)d_in[13]; const float* ee_be = (const float*)d_in[14];
    const float* ra_w1 = (const float*)d_in[15]; const float* ra_b1 = (const float*)d_in[16];
    const float* ra_w2 = (const float*)d_in[17]; const float* ra_b2 = (const float*)d_in[18];
    const float* gat_wl  = (const float*)d_in[19];
    const float* gat_wa  = (const float*)d_in[20];
    const float* gat_we  = (const float*)d_in[21];
    const float* gat_wo  = (const float*)d_in[22];
    const float* gat_wob = (const float*)d_in[23];
    const float* ow_1 = (const float*)d_in[24]; const float* ob_1 = (const float*)d_in[25];
    const float* ow_2 = (const float*)d_in[26]; const float* ob_2 = (const float*)d_in[27];
    const float* o_g  = (const float*)d_in[28]; const float* o_be = (const float*)d_in[29];

    const int N = in_sizes[0] / 9;       // 20000 (multiple of 16)
    const int E = in_sizes[1] / 4;       // 320000 (multiple of 16 and 256)

    float* ws     = (float*)d_ws;
    float* h      = ws;                              // N*64
    float* ef     = h + (size_t)N * 64;              // E*32
    float* rw     = ef + (size_t)E * 32;             // E
    float* xh     = rw + (size_t)E;                  // N*64
    float* xproj  = xh + (size_t)N * 64;             // N*256
    float* sij    = xproj + (size_t)N * 256;         // N*8
    float* palpha = sij + (size_t)N * 8;             // E*4
    float* agg    = palpha + (size_t)E * 4;          // N*256
    float* amax   = agg + (size_t)N * 256;           // 4
    float* asum   = amax + 4;                        // 4
    float* hid1   = xproj;                           // alias (N*128 <= N*256)
    float* outpre = agg;                             // alias (N*256)

    // encoders + risk weights
    k_node_enc<<<N, 64, 0, stream>>>(x, ne_w1, ne_b1, ne_w2, ne_b2, ne_g, ne_be, h, N);
    k_edge_enc<<<(E + 255) / 256, 256, 0, stream>>>(edge_attr, ee_w1, ee_b1, ee_w2, ee_b2,
                                                    ee_g, ee_be, ef, E);
    k_risk<<<E / 16, 64, 0, stream>>>(h, ef, edge_index, ra_w1, ra_b1, ra_w2, ra_b2, rw, E);

    // GAT layers
    for (int l = 0; l < 3; ++l) {
        const float* wl  = gat_wl  + (size_t)l * 64 * 256;
        const float* wa  = gat_wa  + (size_t)l * 160;
        const float* we  = gat_we  + (size_t)l * 32 * 256;
        const float* wo  = gat_wo  + (size_t)l * 64 * 64;
        const float* wob = gat_wob + (size_t)l * 64;
        const float* xin = (l == 0) ? h : xh;

        k_xproj<<<N / 16, 256, 0, stream>>>(xin, wl, wa, xproj, sij, N);
        k_init<<<(N * 256 + 255) / 256, 256, 0, stream>>>(agg, amax, asum, N * 256);
        k_alpha<<<(E + 255) / 256, 256, 0, stream>>>(ef, edge_index, sij, rw, wa, palpha, amax, E);
        k_expsum<<<(E + 255) / 256, 256, 0, stream>>>(palpha, amax, asum, E);
        k_agg<<<E / 16, 128, 0, stream>>>(ef, edge_index, palpha, asum, we, xproj, agg, E);
        k_otrans<<<N, 64, 0, stream>>>(agg, wo, wob, xh, N);
    }

    // output MLP + LayerNorm
    k_out1<<<N / 16, 128, 0, stream>>>(xh, ow_1, ob_1, hid1, N);
    k_out2<<<N / 16, 256, 0, stream>>>(hid1, ow_2, ob_2, outpre, N);
    k_ln256<<<N, 256, 0, stream>>>(outpre, o_g, o_be, (float*)d_out, N);
}